// Devign_simplify_22857815949593
// MI455X (gfx1250) — compile-verified
//
#include <hip/hip_runtime.h>
#include <hip/hip_bf16.h>

// ---------------------------------------------------------------------------
// GGNN (Devign) forward on gfx1250.
// h=[50000,200] fp32.  6 steps of:  m=h@W_t ; agg=scatter_add(m[src],dst) ;
// h=GRU(agg,h).  Then relu -> segment_max over 256 graphs -> 2-way classifier
// -> sigmoid.  GEMMs use V_WMMA_F32_16X16X4_F32 (fp32-exact vs reference).
// B operands are pre-swizzled into per-lane WMMA fragment order so the hot
// loop does exactly one coalesced b64 load per WMMA per lane.
// ---------------------------------------------------------------------------

#define N_NODES   50000
#define N_EDGES   400000
#define N_GRAPHS  256
#define DIM       200
#define N_STEPS   6
#define D3        600
#define ASTR      201              // LDS A-tile row stride (bank-conflict free)
#define D3P       608              // 600 padded to 16 (gi/gh LDS row stride)
#define KSTEPS    (DIM / 4)        // 50 WMMA K-steps
#define CT_M      13               // col tiles covering 208 >= 200
#define CT_G      38               // col tiles covering 608 >= 600
#define ROW_TILES (N_NODES / 16)   // 3125 (exact)

typedef float v2f __attribute__((ext_vector_type(2)));
typedef float v8f __attribute__((ext_vector_type(8)));

__device__ __forceinline__ v8f wmma_f32_16x16x4(v2f a, v2f b, v8f c) {
  // (neg_a, A, neg_b, B, c_mod, C, reuse_a, reuse_b)
  return __builtin_amdgcn_wmma_f32_16x16x4_f32(false, a, false, b, (short)0, c,
                                               false, false);
}

__device__ __forceinline__ float sigmoidf_(float x) {
  return 1.0f / (1.0f + __expf(-x));
}

// ---------------------------------------------------------------------------
// Zero fill
// ---------------------------------------------------------------------------
__global__ void fill0_kernel(float* __restrict__ p, int n) {
  int i = blockIdx.x * 256 + threadIdx.x;
  if (i < n) p[i] = 0.0f;
}

// ---------------------------------------------------------------------------
// Weight prep: swizzle B matrices into per-lane WMMA fragment order.
// Fragment layout (ISA 7.12.2, f32 16x16x4): lane L (grp=L>>4, lm=L&15)
//   b.x = B[k0 + 2*grp    ][ct*16 + lm]
//   b.y = B[k0 + 2*grp + 1][ct*16 + lm]
// Stored as float2 at Bswz[(ct*KSTEPS + ks)*32 + L]  -> one b64 load / WMMA.
//   Wswz  : [6][13][50][32] float2   B = W_t           (col >= 200 -> 0)
//   wihSwz: [38][50][32]    float2   B[k][j] = w_ih[j][k]  (j >= 600 -> 0)
//   whhSwz: [38][50][32]    float2   B[k][j] = w_hh[j][k]
// ---------------------------------------------------------------------------
__global__ void prep_kernel(const float* __restrict__ ggnn_w,
                            const float* __restrict__ w_ih,
                            const float* __restrict__ w_hh,
                            float* __restrict__ Wswz,
                            float* __restrict__ wihSwz,
                            float* __restrict__ whhSwz) {
  int p = blockIdx.x * 256 + threadIdx.x;

  // ggnn weights: 6*13*50*32 fragment pairs
  if (p < N_STEPS * CT_M * KSTEPS * 32) {
    int t   = p / (CT_M * KSTEPS * 32);
    int r   = p % (CT_M * KSTEPS * 32);
    int ct  = r / (KSTEPS * 32);
    int ks  = (r / 32) % KSTEPS;
    int L   = r % 32;
    int col = ct * 16 + (L & 15);
    int k   = ks * 4 + 2 * (L >> 4);
    const float* W = ggnn_w + t * DIM * DIM;
    float bx = (col < DIM) ? W[k * DIM + col] : 0.0f;
    float by = (col < DIM) ? W[(k + 1) * DIM + col] : 0.0f;
    Wswz[2 * p]     = bx;
    Wswz[2 * p + 1] = by;
  }

  // GRU weights: 38*50*32 fragment pairs each
  if (p < CT_G * KSTEPS * 32) {
    int ct  = p / (KSTEPS * 32);
    int ks  = (p / 32) % KSTEPS;
    int L   = p % 32;
    int col = ct * 16 + (L & 15);
    int k   = ks * 4 + 2 * (L >> 4);
    bool ok = (col < D3);
    wihSwz[2 * p]     = ok ? w_ih[col * DIM + k] : 0.0f;
    wihSwz[2 * p + 1] = ok ? w_ih[col * DIM + k + 1] : 0.0f;
    whhSwz[2 * p]     = ok ? w_hh[col * DIM + k] : 0.0f;
    whhSwz[2 * p + 1] = ok ? w_hh[col * DIM + k + 1] : 0.0f;
  }
}

// ---------------------------------------------------------------------------
// Stage a 16xDIM fp32 tile into LDS (stride ASTR) with b128 global loads.
// rows are 800B so every row start is 16B aligned; 50 float4 per row.
// ---------------------------------------------------------------------------
__device__ __forceinline__ void stage_tile(const float* __restrict__ g,
                                           int row0, float* __restrict__ sA,
                                           int tid) {
  for (int i = tid; i < 16 * (DIM / 4); i += 256) {
    int r = i / (DIM / 4), c4 = i % (DIM / 4);
    const float4 v = ((const float4*)(g + (size_t)(row0 + r) * DIM))[c4];
    float* d = sA + r * ASTR + 4 * c4;
    d[0] = v.x; d[1] = v.y; d[2] = v.z; d[3] = v.w;
  }
}

// ---------------------------------------------------------------------------
// m = h @ W_t    (one block per 16-node row tile, 8 waves, wave-per-col-tile)
// ---------------------------------------------------------------------------
__global__ void __launch_bounds__(256)
gemm_m_kernel(const float* __restrict__ h, const float* __restrict__ Wswz,
              float* __restrict__ m, int step) {
  __shared__ float sA[16 * ASTR];
  const int row0 = blockIdx.x * 16;
  stage_tile(h, row0, sA, threadIdx.x);
  __syncthreads();

  const int wave = threadIdx.x >> 5;
  const int lane = threadIdx.x & 31;
  const int grp  = lane >> 4;
  const int lm   = lane & 15;

  for (int ct = wave; ct < CT_M; ct += 8) {
    const int col = ct * 16 + lm;
    const v2f* Bw =
        (const v2f*)Wswz + (size_t)(step * CT_M + ct) * KSTEPS * 32 + lane;
    v8f acc = {0.f, 0.f, 0.f, 0.f, 0.f, 0.f, 0.f, 0.f};
    for (int ks = 0; ks < KSTEPS; ++ks) {
      v2f a;
      a.x = sA[lm * ASTR + 4 * ks + 2 * grp];
      a.y = sA[lm * ASTR + 4 * ks + 2 * grp + 1];
      acc = wmma_f32_16x16x4(a, Bw[ks * 32], acc);
    }
    if (col < DIM) {
#pragma unroll
      for (int v = 0; v < 8; ++v) {                // C row = v + 8*grp
        m[(row0 + v + 8 * grp) * DIM + col] = acc[v];
      }
    }
  }
}

// ---------------------------------------------------------------------------
// Edge scatter: agg[dst] += m[src].  One wave32 per edge; f32 HW atomics hit
// L2 (m and agg are both L2-resident: 40MB each vs 192MB L2).
// ---------------------------------------------------------------------------
__global__ void __launch_bounds__(256)
scatter_kernel(const float* __restrict__ m, const int* __restrict__ src,
               const int* __restrict__ dst, float* __restrict__ agg) {
  int gid  = blockIdx.x * 256 + threadIdx.x;
  int e    = gid >> 5;
  int lane = gid & 31;
  if (e >= N_EDGES) return;
  const int s = src[e], d = dst[e];
  const float* ms = m + (size_t)s * DIM;
  float* ad = agg + (size_t)d * DIM;
  for (int f = lane; f < DIM; f += 32) {
    unsafeAtomicAdd(&ad[f], ms[f]);   // global_atomic_add_f32, no return
  }
}

// ---------------------------------------------------------------------------
// Fused GRU: gi = agg@w_ih^T + b_ih ; gh = h@w_hh^T + b_hh ; gate math ; h'.
// One block per 16-node tile.  gi/gh tiles (16x600) never touch HBM.
// LDS: 2*16*201 + 2*16*608 floats = ~101 KB (<= 320 KB/WGP).
// ---------------------------------------------------------------------------
__global__ void __launch_bounds__(256)
gru_fused_kernel(const float* __restrict__ agg,
                 const float* __restrict__ wihSwz,
                 const float* __restrict__ whhSwz,
                 const float* __restrict__ b_ih, const float* __restrict__ b_hh,
                 float* __restrict__ h) {
  __shared__ float sAagg[16 * ASTR];
  __shared__ float sAh  [16 * ASTR];
  __shared__ float sGI  [16 * D3P];
  __shared__ float sGH  [16 * D3P];
  const int row0 = blockIdx.x * 16;

  stage_tile(agg, row0, sAagg, threadIdx.x);
  stage_tile(h, row0, sAh, threadIdx.x);
  __syncthreads();

  const int wave = threadIdx.x >> 5;
  const int lane = threadIdx.x & 31;
  const int grp  = lane >> 4;
  const int lm   = lane & 15;

  // 76 tile jobs: jobs [0,38) -> gi, [38,76) -> gh.
  for (int job = wave; job < 2 * CT_G; job += 8) {
    const bool is_gh  = (job >= CT_G);
    const int ct      = is_gh ? job - CT_G : job;
    const float* sA   = is_gh ? sAh : sAagg;
    const float* Bsz  = is_gh ? whhSwz : wihSwz;
    const float* bias = is_gh ? b_hh : b_ih;
    float* sOut       = is_gh ? sGH : sGI;
    const int col     = ct * 16 + lm;
    const v2f* Bw     = (const v2f*)Bsz + (size_t)ct * KSTEPS * 32 + lane;
    const float bv    = (col < D3) ? bias[col] : 0.0f;
    v8f acc = {bv, bv, bv, bv, bv, bv, bv, bv};
    for (int ks = 0; ks < KSTEPS; ++ks) {
      v2f a;
      a.x = sA[lm * ASTR + 4 * ks + 2 * grp];
      a.y = sA[lm * ASTR + 4 * ks + 2 * grp + 1];
      acc = wmma_f32_16x16x4(a, Bw[ks * 32], acc);
    }
#pragma unroll
    for (int v = 0; v < 8; ++v) {
      sOut[(v + 8 * grp) * D3P + col] = acc[v];
    }
  }
  __syncthreads();

  // GRU gates + in-place h update (depends only on precomputed gi/gh + old h).
  for (int i = threadIdx.x; i < 16 * DIM; i += 256) {
    int r = i / DIM, j = i % DIM;
    float ir = sGI[r * D3P + j];
    float iz = sGI[r * D3P + j + DIM];
    float in = sGI[r * D3P + j + 2 * DIM];
    float hr = sGH[r * D3P + j];
    float hz = sGH[r * D3P + j + DIM];
    float hn = sGH[r * D3P + j + 2 * DIM];
    float rg = sigmoidf_(ir + hr);
    float zg = sigmoidf_(iz + hz);
    float ng = tanhf(in + rg * hn);
    float ho = sAh[r * ASTR + j];
    h[(row0 + r) * DIM + j] = (1.0f - zg) * ng + zg * ho;
  }
}

// ---------------------------------------------------------------------------
// ReLU + segment-max pool.  relu >= 0 so int-bit atomicMax on zeroed buffer
// is exact float max.
// ---------------------------------------------------------------------------
__global__ void __launch_bounds__(256)
pool_kernel(const float* __restrict__ h, const int* __restrict__ batch,
            float* __restrict__ pooled) {
  int idx = blockIdx.x * 256 + threadIdx.x;
  if (idx >= N_NODES * DIM) return;
  int n = idx / DIM, d = idx % DIM;
  float v = fmaxf(h[idx], 0.0f);
  int g = batch[n];
  atomicMax((int*)&pooled[g * DIM + d], __float_as_int(v));
}

// ---------------------------------------------------------------------------
// Classifier: logits = pooled @ cls_w^T + cls_b ; sigmoid.  512 outputs.
// ---------------------------------------------------------------------------
__global__ void cls_kernel(const float* __restrict__ pooled,
                           const float* __restrict__ cls_w,
                           const float* __restrict__ cls_b,
                           float* __restrict__ out) {
  int idx = blockIdx.x * 256 + threadIdx.x;
  if (idx >= N_GRAPHS * 2) return;
  int g = idx >> 1, c = idx & 1;
  float acc = cls_b[c];
  for (int d = 0; d < DIM; ++d) acc += pooled[g * DIM + d] * cls_w[c * DIM + d];
  out[g * 2 + c] = sigmoidf_(acc);
}

// ---------------------------------------------------------------------------
// Host launch
// ---------------------------------------------------------------------------
extern "C" void kernel_launch(void* const* d_in, const int* in_sizes, int n_in,
                              void* d_out, int out_size, void* d_ws,
                              size_t ws_size, hipStream_t stream) {
  (void)in_sizes; (void)n_in; (void)out_size; (void)ws_size;
  const float* x       = (const float*)d_in[0];
  const int*   eidx    = (const int*)d_in[1];   // [2, E] flat: row0=src, row1=dst
  const int*   batch   = (const int*)d_in[2];
  const float* ggnn_w  = (const float*)d_in[3];
  const float* w_ih    = (const float*)d_in[4];
  const float* w_hh    = (const float*)d_in[5];
  const float* b_ih    = (const float*)d_in[6];
  const float* b_hh    = (const float*)d_in[7];
  const float* cls_w   = (const float*)d_in[8];
  const float* cls_b   = (const float*)d_in[9];
  float* out = (float*)d_out;

  // Workspace layout (floats): ~122 MB total.
  float* ws = (float*)d_ws;
  const size_t ND = (size_t)N_NODES * DIM;                 // 10,000,000
  const size_t WSWZ_N   = (size_t)N_STEPS * CT_M * KSTEPS * 64;  // 249,600
  const size_t GRUSWZ_N = (size_t)CT_G * KSTEPS * 64;            // 121,600
  float* h      = ws;
  float* m      = h + ND;
  float* agg    = m + ND;
  float* Wswz   = agg + ND;
  float* wihSwz = Wswz + WSWZ_N;
  float* whhSwz = wihSwz + GRUSWZ_N;
  float* pooled = whhSwz + GRUSWZ_N;                       // 256*200

  const int prep_n = N_STEPS * CT_M * KSTEPS * 32;         // 124,800 pairs
  prep_kernel<<<(prep_n + 255) / 256, 256, 0, stream>>>(ggnn_w, w_ih, w_hh,
                                                        Wswz, wihSwz, whhSwz);
  fill0_kernel<<<(N_GRAPHS * DIM + 255) / 256, 256, 0, stream>>>(
      pooled, N_GRAPHS * DIM);
  hipMemcpyAsync(h, x, ND * sizeof(float), hipMemcpyDeviceToDevice, stream);

  const int nd_blocks      = (int)((ND + 255) / 256);      // 39063
  const int scatter_blocks = (N_EDGES * 32) / 256;         // 50000

  for (int t = 0; t < N_STEPS; ++t) {
    gemm_m_kernel<<<ROW_TILES, 256, 0, stream>>>(h, Wswz, m, t);
    fill0_kernel<<<nd_blocks, 256, 0, stream>>>(agg, (int)ND);
    scatter_kernel<<<scatter_blocks, 256, 0, stream>>>(m, eidx, eidx + N_EDGES,
                                                       agg);
    gru_fused_kernel<<<ROW_TILES, 256, 0, stream>>>(agg, wihSwz, whhSwz, b_ih,
                                                    b_hh, h);
  }

  pool_kernel<<<nd_blocks, 256, 0, stream>>>(h, batch, pooled);
  cls_kernel<<<(N_GRAPHS * 2 + 255) / 256, 256, 0, stream>>>(pooled, cls_w,
                                                             cls_b, out);
}